// OTAssignerSRFDet_8710193676395
// MI455X (gfx1250) — compile-verified
//
#include <hip/hip_runtime.h>
#include <hip/hip_bf16.h>
#include <math.h>

// Problem constants (from reference)
#define BB 16
#define PP 6000
#define GG 128
#define CC 10
#define CLS_WF 2.0f
#define ALPHAF 0.25f
#define REG_WF 0.25f
#define IOU_WF 0.25f
#define EPSF 1e-12f
#define INF1F 10000.0f
#define INF2F 100000.0f
#define NUM_HEADS_I 6

typedef float v2f __attribute__((ext_vector_type(2)));
typedef float v8f __attribute__((ext_vector_type(8)));

// ---------- device helpers ----------

__device__ __forceinline__ float focal_posneg(float x) {
  // p = sigmoid(x); pos - neg (focal terms)
  float p = 1.0f / (1.0f + __expf(-x));
  float omp = 1.0f - p;
  float pos = -logf(p + EPSF) * ALPHAF * omp * omp;
  float neg = -logf(omp + EPSF) * (1.0f - ALPHAF) * p * p;
  return pos - neg;
}

// pred AABB from pred_boxes[0..7] = {ctr(3), log-size(3), sin, cos}
__device__ __forceinline__ void pred_aabb(const float* pb, float* pmin, float* pmax, float& vp) {
  float dx = __expf(pb[3]), dy = __expf(pb[4]), dz = __expf(pb[5]);
  float s = pb[6], c = pb[7];
  float r = fmaxf(sqrtf(s * s + c * c), 1e-20f);
  float ca = fabsf(c) / r, sa = fabsf(s) / r;     // |cos(atan2(s,c))|, |sin(...)|
  float hx = 0.5f * (dx * ca + dy * sa);
  float hy = 0.5f * (dx * sa + dy * ca);
  float hz = 0.5f * dz;
  pmin[0] = pb[0] - hx; pmax[0] = pb[0] + hx;
  pmin[1] = pb[1] - hy; pmax[1] = pb[1] + hy;
  pmin[2] = pb[2] - hz; pmax[2] = pb[2] + hz;
  vp = 8.0f * hx * hy * hz;
}

// ---------- kernel 0: zero matching + colcnt ----------
__global__ void kZero(unsigned int* m32, int* colcnt) {
  size_t i = (size_t)blockIdx.x * blockDim.x + threadIdx.x;
  const size_t n = (size_t)BB * PP * GG / 4;
  if (i < n) m32[i] = 0u;
  if (i < (size_t)BB * GG) colcnt[i] = 0;
}

// ---------- kernel 1: per-GT precompute ----------
// layout (24 floats): [0:3]=gmin [3:6]=gmax [6]=vg [8:16]=gt_norm8 [16:19]=lo [19:22]=hi
__global__ void kGtPre(const float* __restrict__ gt, float* __restrict__ gtpre) {
  int i = blockIdx.x * blockDim.x + threadIdx.x;
  if (i >= BB * GG) return;
  const float* g = gt + (size_t)i * 7;
  float* o = gtpre + (size_t)i * 24;
  float cx = g[0], cy = g[1], cz = g[2], dx = g[3], dy = g[4], dz = g[5], yaw = g[6];
  float cyw = cosf(yaw), syw = sinf(yaw);
  float ca = fabsf(cyw), sa = fabsf(syw);
  float hx = 0.5f * (dx * ca + dy * sa), hy = 0.5f * (dx * sa + dy * ca), hz = 0.5f * dz;
  o[0] = cx - hx; o[1] = cy - hy; o[2] = cz - hz;
  o[3] = cx + hx; o[4] = cy + hy; o[5] = cz + hz;
  o[6] = 8.0f * hx * hy * hz; o[7] = 0.0f;
  o[8] = cx; o[9] = cy; o[10] = cz;
  o[11] = logf(dx); o[12] = logf(dy); o[13] = logf(dz);
  o[14] = syw; o[15] = cyw;
  o[16] = cx - 1.5f * dx; o[17] = cy - 1.5f * dy; o[18] = cz - 1.5f * dz;
  o[19] = cx + 1.5f * dx; o[20] = cy + 1.5f * dy; o[21] = cz + 1.5f * dz;
  o[22] = 0.0f; o[23] = 0.0f;
}

// ---------- kernel 2: class cost via WMMA (posneg[P x C] * onehot[C x G]) ----------
// One wave per (batch, 16-row P tile); A reused across all 8 G tiles; 3 K-tiles (C=10 -> pad 12).
__global__ void kClsWmma(const float* __restrict__ logits, const int* __restrict__ labels,
                         float* __restrict__ cost) {
  const int PT = PP / 16;                                     // 375 tiles
  int gwave = blockIdx.x * (blockDim.x >> 5) + (threadIdx.x >> 5);
  if (gwave >= BB * PT) return;                               // grid is exact; wave-uniform
  int b = gwave / PT, pt = gwave % PT;
  int lane = threadIdx.x & 31;
  int M = lane & 15;            // A row / B,D column within tile
  int kh = lane >> 4;           // half-wave selects K pair (A/B) and row half (D)

  // A: 16x4 f32, scaled by CLS_W. v.x -> K = 2*kh, v.y -> K = 2*kh+1 (per K-tile)
  const float* lg = logits + ((size_t)b * PP + (size_t)pt * 16 + M) * CC;
  v2f A[3];
#pragma unroll
  for (int t = 0; t < 3; ++t) {
    int c0 = 4 * t + 2 * kh;
    A[t].x = (c0 < CC) ? focal_posneg(lg[c0]) * CLS_WF : 0.0f;
    A[t].y = (c0 + 1 < CC) ? focal_posneg(lg[c0 + 1]) * CLS_WF : 0.0f;
  }

  const int* lab = labels + (size_t)b * GG;
  for (int gt = 0; gt < GG / 16; ++gt) {
    int labN = lab[gt * 16 + M];            // one-hot column for N = lane&15
    v8f acc = {0.f, 0.f, 0.f, 0.f, 0.f, 0.f, 0.f, 0.f};
#pragma unroll
    for (int t = 0; t < 3; ++t) {
      int c0 = 4 * t + 2 * kh;
      v2f Bv;
      Bv.x = (labN == c0) ? 1.0f : 0.0f;
      Bv.y = (labN == c0 + 1) ? 1.0f : 0.0f;
      acc = __builtin_amdgcn_wmma_f32_16x16x4_f32(false, A[t], false, Bv,
                                                  (short)0, acc, false, false);
    }
    // D layout: VGPR r -> row (r + 8*kh), col N = lane&15
    float* cb = cost + ((size_t)b * PP + (size_t)pt * 16) * GG + gt * 16;
#pragma unroll
    for (int r = 0; r < 8; ++r)
      cb[(size_t)(r + kh * 8) * GG + M] = acc[r];
  }
}

// ---------- kernel 3: finalize cost (reg + iou + masks + valid) ----------
__global__ void kCostFinalize(const float* __restrict__ pred, const float* __restrict__ gtpre,
                              float* __restrict__ cost) {
  const int BPB = (PP + 255) / 256;     // 24 blocks per batch
  int b = blockIdx.x / BPB;
  int p = (blockIdx.x % BPB) * 256 + threadIdx.x;
  __shared__ float sg[GG * 24];
  for (int i = threadIdx.x; i < GG * 24; i += 256)
    sg[i] = gtpre[(size_t)b * GG * 24 + i];
  __syncthreads();
  if (p >= PP) return;

  float pb[10];
  const float* src = pred + ((size_t)b * PP + p) * 10;
#pragma unroll
  for (int i = 0; i < 10; ++i) pb[i] = src[i];
  float pmin[3], pmax[3], vp;
  pred_aabb(pb, pmin, pmax, vp);

  // pass 1: in_box & in_ctr masks, validity
  unsigned mask[4] = {0u, 0u, 0u, 0u};
  bool valid = false;
  for (int g = 0; g < GG; ++g) {
    const float* o = sg + g * 24;
    bool inb = pb[0] > o[0] && pb[0] < o[3] && pb[1] > o[1] && pb[1] < o[4] &&
               pb[2] > o[2] && pb[2] < o[5];
    bool inc = pb[0] > o[16] && pb[0] < o[19] && pb[1] > o[17] && pb[1] < o[20] &&
               pb[2] > o[18] && pb[2] < o[21];
    valid = valid || inb || inc;
    if (inb && inc) mask[g >> 5] |= (1u << (g & 31));
  }
  float vpen = valid ? 0.0f : INF1F;

  // pass 2: accumulate into cls-cost already in buffer
  float* crow = cost + ((size_t)b * PP + p) * GG;
  for (int g = 0; g < GG; ++g) {
    const float* o = sg + g * 24;
    float reg = 0.0f;
#pragma unroll
    for (int k = 0; k < 8; ++k) reg += fabsf(pb[k] - o[8 + k]);
    float ix = fminf(pmax[0], o[3]) - fmaxf(pmin[0], o[0]);
    float iy = fminf(pmax[1], o[4]) - fmaxf(pmin[1], o[1]);
    float iz = fminf(pmax[2], o[5]) - fmaxf(pmin[2], o[2]);
    float inter = fmaxf(ix, 0.0f) * fmaxf(iy, 0.0f) * fmaxf(iz, 0.0f);
    float iou = inter / (vp + o[6] - inter + 1e-7f);
    bool inboth = (mask[g >> 5] >> (g & 31)) & 1u;
    crow[g] += reg * REG_WF - iou * IOU_WF + (inboth ? 0.0f : 100.0f) + vpen;
  }
}

// ---------- kernel 4: dynamic-k per (b,g) from top-5 IoUs ----------
__global__ void kDyn(const float* __restrict__ pred, const float* __restrict__ gtpre,
                     const int* __restrict__ headIdx, int* __restrict__ dyn) {
  int bg = blockIdx.x;
  int b = bg / GG;
  const float* o = gtpre + (size_t)bg * 24;
  float g0 = o[0], g1 = o[1], g2 = o[2], g3 = o[3], g4 = o[4], g5 = o[5], vg = o[6];
  int tid = threadIdx.x;
  float top[5] = {-1.f, -1.f, -1.f, -1.f, -1.f};
  for (int p = tid; p < PP; p += 128) {
    const float* src = pred + ((size_t)b * PP + p) * 10;
    float pb[8];
#pragma unroll
    for (int i = 0; i < 8; ++i) pb[i] = src[i];
    float pmin[3], pmax[3], vp;
    pred_aabb(pb, pmin, pmax, vp);
    float ix = fminf(pmax[0], g3) - fmaxf(pmin[0], g0);
    float iy = fminf(pmax[1], g4) - fmaxf(pmin[1], g1);
    float iz = fminf(pmax[2], g5) - fmaxf(pmin[2], g2);
    float inter = fmaxf(ix, 0.0f) * fmaxf(iy, 0.0f) * fmaxf(iz, 0.0f);
    float iou = inter / (vp + vg - inter + 1e-7f);
    if (iou > top[4]) {
      top[4] = iou;
      for (int k = 4; k > 0 && top[k] > top[k - 1]; --k) {
        float t = top[k]; top[k] = top[k - 1]; top[k - 1] = t;
      }
    }
  }
  __shared__ float st[128 * 5];
#pragma unroll
  for (int k = 0; k < 5; ++k) st[tid * 5 + k] = top[k];
  __syncthreads();
  if (tid == 0) {
    float gl[5] = {-1.f, -1.f, -1.f, -1.f, -1.f};
    for (int i = 0; i < 128 * 5; ++i) {
      float v = st[i];
      if (v > gl[4]) {
        gl[4] = v;
        for (int k = 4; k > 0 && gl[k] > gl[k - 1]; --k) {
          float t = gl[k]; gl[k] = gl[k - 1]; gl[k - 1] = t;
        }
      }
    }
    float s = gl[0] + gl[1] + gl[2] + gl[3] + gl[4];
    int d = (int)(s - 0.5f * (float)(NUM_HEADS_I - headIdx[0]));  // trunc toward 0
    if (d < 1) d = 1;
    if (d > 8) d = 8;
    dyn[bg] = d;
  }
}

// ---------- kernel 5: initial matching = (stable rank in column) < dyn ----------
#define BK 8
__global__ void kInitMatch(const float* __restrict__ cost, const int* __restrict__ dyn,
                           unsigned char* __restrict__ matching) {
  int bg = blockIdx.x;
  int b = bg / GG, g = bg % GG;
  const float* cB = cost + (size_t)b * PP * GG + g;
  int tid = threadIdx.x;
  float val[BK]; int idx[BK];
#pragma unroll
  for (int k = 0; k < BK; ++k) { val[k] = 3.0e38f; idx[k] = 0x7fffffff; }
  for (int p = tid; p < PP; p += 128) {
    float v = cB[(size_t)p * GG];
    if (v < val[BK - 1] || (v == val[BK - 1] && p < idx[BK - 1])) {
      val[BK - 1] = v; idx[BK - 1] = p;
      for (int k = BK - 1; k > 0; --k) {
        bool lt = val[k] < val[k - 1] || (val[k] == val[k - 1] && idx[k] < idx[k - 1]);
        if (!lt) break;
        float tv = val[k]; val[k] = val[k - 1]; val[k - 1] = tv;
        int ti = idx[k]; idx[k] = idx[k - 1]; idx[k - 1] = ti;
      }
    }
  }
  __shared__ float sv[128 * BK];
  __shared__ int si[128 * BK];
#pragma unroll
  for (int k = 0; k < BK; ++k) { sv[tid * BK + k] = val[k]; si[tid * BK + k] = idx[k]; }
  __syncthreads();
  if (tid == 0) {
    float gv[BK]; int gi[BK];
#pragma unroll
    for (int k = 0; k < BK; ++k) { gv[k] = 3.0e38f; gi[k] = 0x7fffffff; }
    for (int i = 0; i < 128 * BK; ++i) {
      float v = sv[i]; int ix = si[i];
      if (v < gv[BK - 1] || (v == gv[BK - 1] && ix < gi[BK - 1])) {
        gv[BK - 1] = v; gi[BK - 1] = ix;
        for (int k = BK - 1; k > 0; --k) {
          bool lt = gv[k] < gv[k - 1] || (gv[k] == gv[k - 1] && gi[k] < gi[k - 1]);
          if (!lt) break;
          float tv = gv[k]; gv[k] = gv[k - 1]; gv[k - 1] = tv;
          int ti = gi[k]; gi[k] = gi[k - 1]; gi[k - 1] = ti;
        }
      }
    }
    int d = dyn[bg]; if (d > BK) d = BK;
    for (int k = 0; k < d; ++k)
      if (gi[k] < PP)
        matching[((size_t)b * PP + gi[k]) * GG + g] = 1;
  }
}

// ---------- kernel 6: prior-row one-hot fix, row argmin, counters ----------
__global__ void kRowFix(const float* __restrict__ cost, unsigned char* __restrict__ matching,
                        unsigned char* __restrict__ prior, int* __restrict__ rowargmin,
                        int* __restrict__ rowcnt, float* __restrict__ pen,
                        int* __restrict__ colcnt) {
  int i = blockIdx.x * blockDim.x + threadIdx.x;
  if (i >= BB * PP) return;
  int b = i / PP;
  const float* crow = cost + (size_t)i * GG;
  unsigned char* mrow = matching + (size_t)i * GG;
  float bv = crow[0]; int bi = 0; int s = 0;
  for (int g = 0; g < GG; ++g) {
    float v = crow[g];
    if (v < bv) { bv = v; bi = g; }           // first minimum (jnp.argmin)
    s += mrow[g];
  }
  bool pr = s > 1;
  prior[i] = pr ? 1 : 0;
  rowargmin[i] = bi;
  pen[i] = 0.0f;
  if (pr) {
    for (int g = 0; g < GG; ++g) mrow[g] = 0;
    mrow[bi] = 1;
    rowcnt[i] = 1;
    atomicAdd(&colcnt[b * GG + bi], 1);
  } else {
    rowcnt[i] = s;
    for (int g = 0; g < GG; ++g)
      if (mrow[g]) atomicAdd(&colcnt[b * GG + g], 1);
  }
}

// ---------- kernel 7: sequential fix-up loop, one block per batch ----------
__global__ void __launch_bounds__(1024)
kLoop(const float* __restrict__ cost, unsigned char* __restrict__ matching,
      float* __restrict__ pen, int* __restrict__ rowcnt, int* __restrict__ colcnt,
      const unsigned char* __restrict__ prior, const int* __restrict__ rowargmin) {
  int b = blockIdx.x;
  const float* cB = cost + (size_t)b * PP * GG;
  unsigned char* mB = matching + (size_t)b * PP * GG;
  float* penB = pen + (size_t)b * PP;
  int* rcB = rowcnt + (size_t)b * PP;
  int* ccB = colcnt + (size_t)b * GG;
  const unsigned char* prB = prior + (size_t)b * PP;
  const int* raB = rowargmin + (size_t)b * PP;

  __shared__ int s_pos[GG];
  __shared__ unsigned char s_unm[GG];
  __shared__ int s_any, s_over;
  int tid = threadIdx.x, nt = blockDim.x;

  for (int it = 0; it < GG + 8; ++it) {
    if (tid == 0) { s_any = 0; s_over = 0; }
    __syncthreads();
    for (int g = tid; g < GG; g += nt) {
      int unm = (ccB[g] == 0);
      s_unm[g] = (unsigned char)unm;
      if (unm) s_any = 1;                    // benign race
    }
    __syncthreads();
    if (!s_any) break;

    // penalties for currently-matched rows (state from loop entry)
    for (int p = tid; p < PP; p += nt)
      if (rcB[p] > 0) penB[p] += INF2F;
    __syncthreads();

    // per-unmatched-column argmin over cost + penalty (wave32-parallel)
    int wave = tid >> 5, lane = tid & 31, nw = nt >> 5;
    for (int g = wave; g < GG; g += nw) {
      if (!s_unm[g]) continue;
      float bv = 3.0e38f; int bi = PP;
      for (int p = lane; p < PP; p += 32) {
        float v = cB[(size_t)p * GG + g] + penB[p];
        if (v < bv || (v == bv && p < bi)) { bv = v; bi = p; }
      }
      for (int off = 16; off > 0; off >>= 1) {
        float ov = __shfl_down(bv, off);
        int oi = __shfl_down(bi, off);
        if (ov < bv || (ov == bv && oi < bi)) { bv = ov; bi = oi; }
      }
      if (lane == 0) s_pos[g] = bi;
    }
    __syncthreads();

    // assign unmatched columns
    for (int g = tid; g < GG; g += nt) {
      if (s_unm[g]) {
        int p = s_pos[g];
        mB[(size_t)p * GG + g] = 1;
        atomicAdd(&rcB[p], 1);
        ccB[g] = 1;
      }
    }
    __syncthreads();

    // over-assignment check
    for (int p = tid; p < PP; p += nt)
      if (rcB[p] > 1) s_over = 1;            // benign race
    __syncthreads();

    // reset prior rows to their one-hot argmin (penalty uniform per row => argmin fixed)
    if (s_over) {
      for (int p = tid; p < PP; p += nt) {
        if (prB[p]) {
          for (int g = 0; g < GG; ++g) {
            if (mB[(size_t)p * GG + g]) {
              mB[(size_t)p * GG + g] = 0;
              atomicSub(&ccB[g], 1);
            }
          }
          int g0 = raB[p];
          mB[(size_t)p * GG + g0] = 1;
          atomicAdd(&ccB[g0], 1);
          rcB[p] = 1;
        }
      }
    }
    __syncthreads();
  }
}

// ---------- kernel 8: outputs (fg as 0/1 float, matched index as float) ----------
__global__ void kOut(const unsigned char* __restrict__ matching, float* __restrict__ out) {
  int i = blockIdx.x * blockDim.x + threadIdx.x;
  if (i >= BB * PP) return;
  const unsigned char* mrow = matching + (size_t)i * GG;
  int m = -1;
  for (int g = 0; g < GG; ++g) {
    if (mrow[g]) { m = g; break; }           // argmax of 0/1 row = first 1
  }
  out[i] = (m >= 0) ? 1.0f : 0.0f;
  out[(size_t)BB * PP + i] = (float)m;
}

// ---------- host ----------
extern "C" void kernel_launch(void* const* d_in, const int* in_sizes, int n_in,
                              void* d_out, int out_size, void* d_ws, size_t ws_size,
                              hipStream_t stream) {
  const float* pred_logits = (const float*)d_in[0];
  const float* pred_boxes = (const float*)d_in[1];
  const float* gt_boxes = (const float*)d_in[2];
  const int* gt_labels = (const int*)d_in[3];
  const int* head_idx = (const int*)d_in[4];
  float* out = (float*)d_out;

  char* ws = (char*)d_ws;
  size_t off = 0;
  auto alloc = [&](size_t bytes) -> void* {
    void* p = ws + off;
    off = (off + bytes + 255) & ~(size_t)255;
    return p;
  };
  float* cost = (float*)alloc((size_t)BB * PP * GG * 4);          // 49.2 MB (fits L2)
  unsigned char* matching = (unsigned char*)alloc((size_t)BB * PP * GG);
  float* gtpre = (float*)alloc((size_t)BB * GG * 24 * 4);
  int* dyn = (int*)alloc((size_t)BB * GG * 4);
  unsigned char* prior = (unsigned char*)alloc((size_t)BB * PP);
  int* rowargmin = (int*)alloc((size_t)BB * PP * 4);
  int* rowcnt = (int*)alloc((size_t)BB * PP * 4);
  float* pen = (float*)alloc((size_t)BB * PP * 4);
  int* colcnt = (int*)alloc((size_t)BB * GG * 4);

  const int zn = (BB * PP * GG / 4 + 255) / 256;
  kZero<<<zn, 256, 0, stream>>>((unsigned int*)matching, colcnt);
  kGtPre<<<(BB * GG + 255) / 256, 256, 0, stream>>>(gt_boxes, gtpre);
  kClsWmma<<<(BB * (PP / 16)) / 8, 256, 0, stream>>>(pred_logits, gt_labels, cost);
  kCostFinalize<<<BB * ((PP + 255) / 256), 256, 0, stream>>>(pred_boxes, gtpre, cost);
  kDyn<<<BB * GG, 128, 0, stream>>>(pred_boxes, gtpre, head_idx, dyn);
  kInitMatch<<<BB * GG, 128, 0, stream>>>(cost, dyn, matching);
  kRowFix<<<(BB * PP + 255) / 256, 256, 0, stream>>>(cost, matching, prior, rowargmin,
                                                     rowcnt, pen, colcnt);
  kLoop<<<BB, 1024, 0, stream>>>(cost, matching, pen, rowcnt, colcnt, prior, rowargmin);
  kOut<<<(BB * PP + 255) / 256, 256, 0, stream>>>(matching, out);
}